// InverseSplat_87943750353186
// MI455X (gfx1250) — compile-verified
//
#include <hip/hip_runtime.h>

typedef float v2f __attribute__((ext_vector_type(2)));
typedef float v8f __attribute__((ext_vector_type(8)));

#define NPIX 4320      // 6*20*36 pixels
#define HW   720       // 20*36
#define CIN  256
#define OC   424       // DBINS + CTX = 64 + 360
#define OSTR 432       // padded to 27 tiles of 16
#define DB   64
#define CTXC 360
#define FHv  20
#define FWv  36
#define MT   2
#define NPTS 276480    // 6*64*20*36 frustum points

// ---------------- K1: logits = feat @ W^T + b  via V_WMMA_F32_16X16X4_F32 ----
// One wave per 16x16 output tile; K=256 in steps of 4.
// Padding columns (ocol >= 424) read a clamped (valid) W row unconditionally;
// their results land in D columns that are never stored, so no predication is
// needed in the inner loop and EXEC stays all-ones around every WMMA.
__global__ __launch_bounds__(256)
void wmma_gemm_logits(const float* __restrict__ feat,   // [6][256][720]
                      const float* __restrict__ Wd,     // [424][256]
                      const float* __restrict__ bd,     // [424]
                      float* __restrict__ logits)       // [4320][432]
{
    const int NT_N = OSTR / 16;                          // 27
    const int NT   = (NPIX / 16) * NT_N;                 // 270*27
    int wid = blockIdx.x * (blockDim.x >> 5) + (threadIdx.x >> 5);
    if (wid >= NT) return;                               // wave-uniform
    int mt = wid / NT_N, nt = wid % NT_N;
    int lane = threadIdx.x & 31;
    int half = lane >> 4;                                // K-pair select
    int l16  = lane & 15;

    // A operand: row = pixel (M); lanes 0-15 carry K=k,k+1, lanes 16-31 K=k+2,k+3.
    // Across lanes each load touches 2 contiguous 64B segments -> coalesced b32.
    int pix = mt * 16 + l16;
    int n   = pix / HW;
    int hw  = pix % HW;
    const float* fb = feat + (size_t)n * CIN * HW + hw;  // + c*HW

    // B operand: col = output channel (N); clamp padded lanes to a valid row.
    int ocol = nt * 16 + l16;
    bool bval = ocol < OC;
    int orow  = bval ? ocol : (OC - 1);
    const float* wb = Wd + (size_t)orow * CIN;           // contiguous over c

    // Fold bias into the accumulator (bias depends only on N = this lane).
    float bv = bval ? bd[ocol] : 0.0f;
    v8f acc = {bv, bv, bv, bv, bv, bv, bv, bv};

    #pragma unroll 8
    for (int k = 0; k < CIN; k += 4) {
        int ka = k + 2 * half;                           // even
        v2f a, b;
        a.x = fb[(size_t)(ka    ) * HW];
        a.y = fb[(size_t)(ka + 1) * HW];
        float2 bw = *(const float2*)(wb + ka);           // global_load_b64
        b.x = bw.x;
        b.y = bw.y;
        acc = __builtin_amdgcn_wmma_f32_16x16x4_f32(false, a, false, b,
                                                    (short)0, acc, false, false);
    }

    // C/D layout: VGPR r -> M=r (lanes 0-15) / M=8+r (lanes 16-31), N = lane&15
    #pragma unroll
    for (int r = 0; r < 8; ++r) {
        int row  = half ? (8 + r) : r;
        int prow = mt * 16 + row;
        if (bval)
            logits[(size_t)prow * OSTR + ocol] = acc[r];
    }
}

// ---------------- K2: softmax over 64 depth bins per pixel (1 wave / pixel) --
__global__ __launch_bounds__(256)
void softmax_depth(float* __restrict__ logits)
{
    int wid = blockIdx.x * (blockDim.x >> 5) + (threadIdx.x >> 5);
    if (wid >= NPIX) return;
    int lane = threadIdx.x & 31;
    float* row = logits + (size_t)wid * OSTR;
    float x0 = row[lane], x1 = row[lane + 32];
    float mx = fmaxf(x0, x1);
    #pragma unroll
    for (int off = 16; off > 0; off >>= 1)
        mx = fmaxf(mx, __shfl_xor(mx, off, 32));
    float e0 = expf(x0 - mx), e1 = expf(x1 - mx);
    float s = e0 + e1;
    #pragma unroll
    for (int off = 16; off > 0; off >>= 1)
        s += __shfl_xor(s, off, 32);
    float inv = 1.0f / s;
    row[lane]      = e0 * inv;
    row[lane + 32] = e1 * inv;
}

// ---------------- K3: fused project + scatter-add splat ---------------------
__device__ inline void inv3x3(const float* __restrict__ K, float* __restrict__ iv)
{
    float a=K[0],b=K[1],c=K[2],d=K[3],e=K[4],f=K[5],g=K[6],h=K[7],i=K[8];
    float A =  (e*i - f*h), B = -(d*i - f*g), C =  (d*h - e*g);
    float r = 1.0f / (a*A + b*B + c*C);
    iv[0] = A*r;            iv[1] = -(b*i - c*h)*r; iv[2] =  (b*f - c*e)*r;
    iv[3] = B*r;            iv[4] =  (a*i - c*g)*r; iv[5] = -(a*f - c*d)*r;
    iv[6] = C*r;            iv[7] = -(a*h - b*g)*r; iv[8] =  (a*e - b*d)*r;
}

__global__ __launch_bounds__(256)
void splat_kernel(const float* __restrict__ pc,    // [4320][432] probs|ctx
                  const float* __restrict__ Ksrc,  // [6][3][3]
                  const float* __restrict__ Esrc,  // [6][4][4]
                  const float* __restrict__ Ktgt,  // [2][3][3]
                  const float* __restrict__ Etgt,  // [2][4][4]
                  float* __restrict__ out)         // [2][360][720]
{
    int wid = blockIdx.x * (blockDim.x >> 5) + (threadIdx.x >> 5);
    if (wid >= NPTS * MT) return;                  // wave-uniform
    int lane = threadIdx.x & 31;
    int m   = wid & 1;
    int pid = wid >> 1;
    int w = pid % FWv; int t = pid / FWv;
    int h = t % FHv;   t /= FHv;
    int d = t % DB;    int n = t / DB;

    float u  = (float)w * (575.0f / 35.0f);        // linspace(0, IW-1, FW)
    float v  = (float)h * (319.0f / 19.0f);        // linspace(0, IH-1, FH)
    float dd = 1.0f + (float)d * 0.921875f;

    float Ki[9];
    inv3x3(Ksrc + n * 9, Ki);
    float px = u * dd, py = v * dd, pz = dd;
    float c0 = Ki[0]*px + Ki[1]*py + Ki[2]*pz;
    float c1 = Ki[3]*px + Ki[4]*py + Ki[5]*pz;
    float c2 = Ki[6]*px + Ki[7]*py + Ki[8]*pz;

    const float* E = Esrc + n * 16;
    float e0 = E[0]*c0 + E[1]*c1 + E[2] *c2 + E[3];
    float e1 = E[4]*c0 + E[5]*c1 + E[6] *c2 + E[7];
    float e2 = E[8]*c0 + E[9]*c1 + E[10]*c2 + E[11];

    const float* Et = Etgt + m * 16;
    float g0 = Et[0]*e0 + Et[1]*e1 + Et[2] *e2 + Et[3];
    float g1 = Et[4]*e0 + Et[5]*e1 + Et[6] *e2 + Et[7];
    float g2 = Et[8]*e0 + Et[9]*e1 + Et[10]*e2 + Et[11];

    float z = fmaxf(g2, 0.1f);
    const float* Kt = Ktgt + m * 9;
    float ut = Kt[0] * g0 / z + Kt[2];
    float vt = Kt[4] * g1 / z + Kt[5];
    int uf = (int)(ut * 0.0625f);                  // TW/IW = TH/IH = 1/16
    int vf = (int)(vt * 0.0625f);                  // trunc-to-zero == astype(i32)
    bool valid = (g2 > 0.1f) & (uf >= 0) & (uf < FWv) & (vf >= 0) & (vf < FHv);
    if (!valid) return;                            // invalid adds exact zero
    int idx = vf * FWv + uf;

    int p = n * HW + h * FWv + w;
    const float* row = pc + (size_t)p * OSTR;
    float prob = row[d];
    float* ob = out + (size_t)m * CTXC * HW + idx;
    for (int ch = lane; ch < CTXC; ch += 32)
        atomicAdd(ob + (size_t)ch * HW, prob * row[DB + ch]);
}

extern "C" void kernel_launch(void* const* d_in, const int* in_sizes, int n_in,
                              void* d_out, int out_size, void* d_ws, size_t ws_size,
                              hipStream_t stream)
{
    const float* feat = (const float*)d_in[0];
    const float* Ksrc = (const float*)d_in[1];
    const float* Esrc = (const float*)d_in[2];
    const float* Ktgt = (const float*)d_in[3];
    const float* Etgt = (const float*)d_in[4];
    const float* Wd   = (const float*)d_in[5];
    const float* bd   = (const float*)d_in[6];
    float* out = (float*)d_out;
    float* lg  = (float*)d_ws;                      // 4320*432*4 B ~= 7.1 MB

    hipMemsetAsync(out, 0, (size_t)out_size * sizeof(float), stream);

    const int tiles = (NPIX / 16) * (OSTR / 16);    // 7290 waves
    wmma_gemm_logits<<<dim3((tiles + 7) / 8), dim3(256), 0, stream>>>(feat, Wd, bd, lg);

    softmax_depth<<<dim3((NPIX + 7) / 8), dim3(256), 0, stream>>>(lg);

    const long long waves = (long long)NPTS * MT;   // 552960 waves
    splat_kernel<<<dim3((unsigned)((waves + 7) / 8)), dim3(256), 0, stream>>>(
        lg, Ksrc, Esrc, Ktgt, Etgt, out);
}